// IntraDecoderAttention_68350109549174
// MI455X (gfx1250) — compile-verified
//
#include <hip/hip_runtime.h>

typedef __attribute__((ext_vector_type(2))) float v2f;
typedef __attribute__((ext_vector_type(4))) float v4f;
typedef __attribute__((ext_vector_type(8))) float v8f;

#define B_      64
#define T_      512
#define H2      1024
#define GSPLIT  8                      // T-splits per batch for the flash pass
#define WPB     8                      // waves per block (256 threads, wave32)
#define ROWS_PER_BLOCK (T_ / GSPLIT)   // 64
#define ROWS_PER_WAVE  (ROWS_PER_BLOCK / WPB) // 8
#define PART_STRIDE 1040               // floats per partial record (1024 acc + m + l + pad)

// ---------------------------------------------------------------------------
// Kernel 1: q = h_d_t (64x1024) @ W (1024x1024), full f32 via WMMA 16x16x4.
// Each wave computes four adjacent 16x16 N-tiles (4 independent WMMA chains).
// A layout (16x4 f32): lanes 0-15 -> M=lane, v0=K0,v1=K1; lanes 16-31 -> K2,K3.
// B layout (4x16 f32): v0: lanes0-15 K=0 row, lanes16-31 K=2 row; v1: K=1/K=3.
// C/D layout (16x16 f32): VGPR v, lanes0-15 -> M=v, N=lane; lanes16-31 -> M=v+8.
// ---------------------------------------------------------------------------
__global__ __launch_bounds__(256) void qgemm_kernel(
    const float* __restrict__ h, const float* __restrict__ W,
    float* __restrict__ q)
{
  const int wave = threadIdx.x >> 5;
  const int lane = threadIdx.x & 31;
  const int wid  = blockIdx.x * WPB + wave;   // 0..255
  const int tm   = wid >> 4;                  // 0..15  (M tile of 16, M=64)
  const int tn   = (wid & 15) * 4;            // base N tile; wave does tn..tn+3
  const int mn   = lane & 15;                 // row (A) / col (B,C) index
  const int kb   = (lane >> 4) * 2;           // 0 or 2

  const float* hrow  = h + (size_t)(tm * 16 + mn) * H2;
  const float* wbase = W + (size_t)tn * 16 + mn;

  v8f acc0 = {}, acc1 = {}, acc2 = {}, acc3 = {};
  for (int k = 0; k < H2; k += 4) {
    v2f a;
    a.x = hrow[k + kb];
    a.y = hrow[k + kb + 1];
    const float* w0 = wbase + (size_t)(k + kb) * H2;
    const float* w1 = wbase + (size_t)(k + kb + 1) * H2;
    v2f b0; b0.x = w0[0];  b0.y = w1[0];
    v2f b1; b1.x = w0[16]; b1.y = w1[16];
    v2f b2; b2.x = w0[32]; b2.y = w1[32];
    v2f b3; b3.x = w0[48]; b3.y = w1[48];
    acc0 = __builtin_amdgcn_wmma_f32_16x16x4_f32(false, a, false, b0, (short)0, acc0, false, false);
    acc1 = __builtin_amdgcn_wmma_f32_16x16x4_f32(false, a, false, b1, (short)0, acc1, false, false);
    acc2 = __builtin_amdgcn_wmma_f32_16x16x4_f32(false, a, false, b2, (short)0, acc2, false, false);
    acc3 = __builtin_amdgcn_wmma_f32_16x16x4_f32(false, a, false, b3, (short)0, acc3, false, false);
  }

  const int mb = (lane >> 4) * 8;
  #pragma unroll
  for (int v = 0; v < 8; ++v) {
    float* qrow = q + (size_t)(tm * 16 + mb + v) * H2 + (size_t)tn * 16 + mn;
    qrow[0]  = acc0[v];
    qrow[16] = acc1[v];
    qrow[32] = acc2[v];
    qrow[48] = acc3[v];
  }
}

// ---------------------------------------------------------------------------
// Kernel 2: single fused pass over prev_h_dec.
//   - copies each row into new_prev (non-temporal stores)
//   - score = q[b] . prev[b,t]  (wave-wide dot, shfl_xor reduce)
//   - online softmax: per-wave running (m, l, acc[1024] in 32 VGPRs/lane)
//   - waves merged in LDS; block writes partial (m, l, acc) to scratch.
// grid = B*GSPLIT blocks x 256 threads.
// ---------------------------------------------------------------------------
__global__ __launch_bounds__(256) void flash_kernel(
    const float* __restrict__ prev, const float* __restrict__ q,
    float* __restrict__ newprev, float* __restrict__ part)
{
  __shared__ float sm_acc[WPB * H2];   // 32 KB
  __shared__ float sm_m[WPB];
  __shared__ float sm_l[WPB];

  const int b    = blockIdx.x / GSPLIT;
  const int g    = blockIdx.x % GSPLIT;
  const int wave = threadIdx.x >> 5;
  const int lane = threadIdx.x & 31;

  // preload q[b] : 8 float4 per lane (lane-interleaved, coalesced)
  const v4f* q4 = (const v4f*)(q + (size_t)b * H2);
  v4f qv[8];
  #pragma unroll
  for (int j = 0; j < 8; ++j) qv[j] = q4[j * 32 + lane];

  float mw = -__builtin_inff();
  float lw = 0.f;
  v4f acc[8] = {};

  const int tbase = g * ROWS_PER_BLOCK + wave * ROWS_PER_WAVE;
  for (int i = 0; i < ROWS_PER_WAVE; ++i) {
    const int t = tbase + i;
    const v4f* row = (const v4f*)(prev + ((size_t)b * T_ + t) * H2);
    v4f*      dst = (v4f*)(newprev + ((size_t)b * (T_ + 1) + t) * H2);

    v4f x[8];
    #pragma unroll
    for (int j = 0; j < 8; ++j) x[j] = row[j * 32 + lane];

    float s = 0.f;
    #pragma unroll
    for (int j = 0; j < 8; ++j) {
      __builtin_nontemporal_store(x[j], &dst[j * 32 + lane]);   // fused concat copy
      s += x[j].x * qv[j].x + x[j].y * qv[j].y
         + x[j].z * qv[j].z + x[j].w * qv[j].w;
    }
    #pragma unroll
    for (int off = 16; off >= 1; off >>= 1)
      s += __shfl_xor(s, off, 32);

    // online softmax update
    const float mnew  = fmaxf(mw, s);
    const float scale = __expf(mw - mnew);   // 0 on first iteration (mw = -inf)
    const float p     = __expf(s - mnew);
    lw = lw * scale + p;
    #pragma unroll
    for (int j = 0; j < 8; ++j)
      acc[j] = acc[j] * scale + p * x[j];
    mw = mnew;
  }

  // per-wave results -> LDS
  v4f* sa = (v4f*)(sm_acc + wave * H2);
  #pragma unroll
  for (int j = 0; j < 8; ++j) sa[j * 32 + lane] = acc[j];
  if (lane == 0) { sm_m[wave] = mw; sm_l[wave] = lw; }
  __syncthreads();

  // merge 8 waves; each thread owns one float4 of the 1024 dims
  float M = -__builtin_inff();
  #pragma unroll
  for (int w = 0; w < WPB; ++w) M = fmaxf(M, sm_m[w]);
  float L = 0.f;
  v4f ct = {};
  #pragma unroll
  for (int w = 0; w < WPB; ++w) {
    const float e = __expf(sm_m[w] - M);
    L += e * sm_l[w];
    ct += e * ((const v4f*)(sm_acc + w * H2))[threadIdx.x];
  }
  float* pp = part + (size_t)(b * GSPLIT + g) * PART_STRIDE;
  ((v4f*)pp)[threadIdx.x] = ct;
  if (threadIdx.x == 0) { pp[H2] = M; pp[H2 + 1] = L; }
}

// ---------------------------------------------------------------------------
// Kernel 3: merge the GSPLIT partials per batch (exact log-sum-exp algebra),
// write ct_d, and write the appended new_prev[b, T, :] = h_d_t[b].
// grid = B blocks x 256 threads.
// ---------------------------------------------------------------------------
__global__ __launch_bounds__(256) void combine_kernel(
    const float* __restrict__ part, const float* __restrict__ h,
    float* __restrict__ out_ct, float* __restrict__ newprev)
{
  const int b   = blockIdx.x;
  const int tid = threadIdx.x;

  float M = -__builtin_inff();
  #pragma unroll
  for (int g = 0; g < GSPLIT; ++g)
    M = fmaxf(M, part[(size_t)(b * GSPLIT + g) * PART_STRIDE + H2]);

  float L = 0.f;
  v4f ct = {};
  #pragma unroll
  for (int g = 0; g < GSPLIT; ++g) {
    const float* pp = part + (size_t)(b * GSPLIT + g) * PART_STRIDE;
    const float e = __expf(pp[H2] - M);
    L  += e * pp[H2 + 1];
    ct += e * ((const v4f*)pp)[tid];
  }
  const float inv = 1.f / L;
  ct *= inv;
  ((v4f*)(out_ct + (size_t)b * H2))[tid] = ct;

  // appended last row of new_prev
  const v4f hv = ((const v4f*)(h + (size_t)b * H2))[tid];
  __builtin_nontemporal_store(
      hv, (v4f*)(newprev + ((size_t)b * (T_ + 1) + T_) * H2) + tid);
}

// ---------------------------------------------------------------------------
extern "C" void kernel_launch(void* const* d_in, const int* in_sizes, int n_in,
                              void* d_out, int out_size, void* d_ws, size_t ws_size,
                              hipStream_t stream) {
  const float* h    = (const float*)d_in[0];   // (B, 2H)
  const float* prev = (const float*)d_in[1];   // (B, T, 2H)
  const float* W    = (const float*)d_in[2];   // (1, 2H, 2H)

  float* out         = (float*)d_out;
  float* out_ct      = out;                          // B*2H floats
  float* out_newprev = out + (size_t)B_ * H2;        // B*(T+1)*2H floats

  float* q    = (float*)d_ws;                        // B*2H floats (256 KB)
  float* part = q + (size_t)B_ * H2;                 // B*GSPLIT*PART_STRIDE floats

  qgemm_kernel  <<<B_ * 16 / (4 * WPB) /* = 32 */, 256, 0, stream>>>(h, W, q);
  flash_kernel  <<<B_ * GSPLIT,                    256, 0, stream>>>(prev, q, out_newprev, part);
  combine_kernel<<<B_,                             256, 0, stream>>>(part, h, out_ct, out_newprev);
}